// PUNet_30468497997876
// MI455X (gfx1250) — compile-verified
//
#include <hip/hip_runtime.h>

typedef float v2f __attribute__((ext_vector_type(2)));
typedef float v8f __attribute__((ext_vector_type(8)));

__device__ __forceinline__ int imin(int a, int b) { return a < b ? a : b; }

// ---------------------------------------------------------------------------
// Weight packing: W(K x C) row-major -> W2[(k/4)*C*4 + c*4 + (k&3)], K padded
// to a multiple of 4 with zeros. Makes the two B-operand elements a lane needs
// (rows ka, ka+1 at one column) contiguous -> single global_load_b64.
// ---------------------------------------------------------------------------
__global__ void pack_w_kernel(const float* __restrict__ W, float* __restrict__ W2,
                              int K, int C, int total /* KP4*C */) {
  const int t = blockIdx.x * blockDim.x + threadIdx.x;
  if (t >= total) return;
  const int k = t / C, c = t % C;
  const float v = (k < K) ? W[k * C + c] : 0.0f;
  W2[((k >> 2) * C + c) * 4 + (k & 3)] = v;
}

// ---------------------------------------------------------------------------
// One-wave 16-row x C layer: out = relu(in * W2 + bias), guard-free K loop.
// A (16x4 fp32): lanes 0-15 -> M=lane, K=(0,1); lanes 16-31 -> K=(2,3)
// B (4x16):      VGPR0: K=0 / K=2 ; VGPR1: K=1 / K=3   (packed layout)
// D (16x16):     VGPR r: lanes 0-15 -> M=r; lanes 16-31 -> M=r+8; N=lane&15
// ---------------------------------------------------------------------------
template <int KP, int SIN, int C, int SOUT>
__device__ __forceinline__ void sa_layer(const float* __restrict__ in,
                                         const float* __restrict__ W2,
                                         const float* __restrict__ bias,
                                         float* __restrict__ outb,
                                         int wv, int lane) {
  const int l15 = lane & 15;
  const int khalf = (lane >> 4) << 1;  // 0 or 2
  const float* arow = in + (wv * 16 + l15) * SIN + khalf;
  const int rbase = wv * 16 + ((lane >> 4) << 3);
  for (int ct = 0; ct < C; ct += 16) {
    const int col = ct + l15;
    const float* wp = W2 + col * 4 + khalf;
    v8f acc = {};
#pragma unroll 4
    for (int kk = 0; kk < KP; kk += 4) {
      const v2f a = *(const v2f*)(arow + kk);  // ds_load_b64 (even stride/offset)
      const v2f b = *(const v2f*)(wp);         // global_load_b64
      wp += 4 * C;
      acc = __builtin_amdgcn_wmma_f32_16x16x4_f32(false, a, false, b,
                                                  (short)0, acc, false, false);
    }
    const float bv = bias[col];
#pragma unroll
    for (int r = 0; r < 8; ++r) {
      outb[(rbase + r) * SOUT + col] = fmaxf(acc[r] + bv, 0.0f);
    }
  }
}

// Fused SA stage: group of 32 samples -> grouped features (zero-padded K) in
// LDS -> 3-layer WMMA MLP (2 waves, one 16-row tile each) -> max-pool.
template <int CIN, int C1, int C2, int C3, int CPTS>
__global__ void sa_fused_kernel(
    const float* __restrict__ xyz, const float* __restrict__ pts,
    const float* __restrict__ newxyz, const int* __restrict__ gidx,
    const float* __restrict__ w1, const float* __restrict__ b1,
    const float* __restrict__ w2, const float* __restrict__ b2,
    const float* __restrict__ w3, const float* __restrict__ b3,
    float* __restrict__ outp, int N, int M) {
  constexpr int KP1 = (CIN + 3) & ~3;
  constexpr int SA = (((KP1 > C2) ? KP1 : C2) + 2) & ~1;  // even, non-pow2
  constexpr int SB = (((C1 > C3) ? C1 : C3) + 2) & ~1;
  extern __shared__ float sm[];
  float* bufA = sm;                 // 32 x SA
  float* bufB = sm + 32 * SA;       // 32 x SB
  const int g = blockIdx.x;         // b*M + m
  const int b = g / M;
  const int tid = threadIdx.x;      // 0..63

  const int* gi = gidx + (long)g * 32;
  const float nx = newxyz[g * 3 + 0];
  const float ny = newxyz[g * 3 + 1];
  const float nz = newxyz[g * 3 + 2];
  for (int e = tid; e < 32 * KP1; e += 64) {
    const int s = e / KP1, c = e % KP1;
    float v = 0.0f;
    if (c < CIN) {
      const int p = gi[s];
      if (c < 3) v = xyz[((long)b * N + p) * 3 + c] - (c == 0 ? nx : (c == 1 ? ny : nz));
      else       v = pts[((long)b * N + p) * CPTS + (c - 3)];
    }
    bufA[s * SA + c] = v;
  }
  __syncthreads();

  const int wv = tid >> 5, lane = tid & 31;
  sa_layer<KP1, SA, C1, SB>(bufA, w1, b1, bufB, wv, lane);
  __syncthreads();
  sa_layer<C1, SB, C2, SA>(bufB, w2, b2, bufA, wv, lane);
  __syncthreads();
  sa_layer<C2, SA, C3, SB>(bufA, w3, b3, bufB, wv, lane);
  __syncthreads();

  for (int c = tid; c < C3; c += 64) {
    float mx = bufB[c];
#pragma unroll 4
    for (int s = 1; s < 32; ++s) mx = fmaxf(mx, bufB[s * SB + c]);
    outp[(long)g * C3 + c] = mx;
  }
}

// ---------------------------------------------------------------------------
// Generic WMMA GEMM: out = act(A(R x K, row stride lda) * W2 + bias).
// One wave computes 16 rows x 64 cols (4 tiles share each A load).
// KP (= lda-compatible padded K) is a multiple of 4; A rows are zero-padded
// so the K loop is guard-free. Columns are clamped for loads, guarded for
// stores. Optional row remap: phys = (row/obr)*obs + row%obr when obs != 0.
// ---------------------------------------------------------------------------
template <int ACT>
__global__ void gemm_wmma_kernel(const float* __restrict__ A,
                                 const float* __restrict__ W2,
                                 const float* __restrict__ bias,
                                 float* __restrict__ out,
                                 int R, int KP, int lda, int C,
                                 int obr, int obs) {
  const int lane = threadIdx.x;  // 0..31
  const int l15 = lane & 15;
  const int khalf = (lane >> 4) << 1;
  const int row = blockIdx.x * 16 + l15;
  const int cbase = blockIdx.y * 64;
  const float* arow = A + (long)row * lda + khalf;

  const int c0 = imin(cbase + 0  + l15, C - 1);
  const int c1 = imin(cbase + 16 + l15, C - 1);
  const int c2 = imin(cbase + 32 + l15, C - 1);
  const int c3 = imin(cbase + 48 + l15, C - 1);
  const float* wp0 = W2 + c0 * 4 + khalf;
  const float* wp1 = W2 + c1 * 4 + khalf;
  const float* wp2 = W2 + c2 * 4 + khalf;
  const float* wp3 = W2 + c3 * 4 + khalf;
  const int wstep = 4 * C;

  v8f a0 = {}, a1 = {}, a2 = {}, a3 = {};
#pragma unroll 2
  for (int kk = 0; kk < KP; kk += 4) {
    const v2f a = *(const v2f*)(arow + kk);  // even offset -> b64
    const v2f b0 = *(const v2f*)wp0; wp0 += wstep;
    const v2f b1 = *(const v2f*)wp1; wp1 += wstep;
    const v2f b2 = *(const v2f*)wp2; wp2 += wstep;
    const v2f b3 = *(const v2f*)wp3; wp3 += wstep;
    a0 = __builtin_amdgcn_wmma_f32_16x16x4_f32(false, a, false, b0, (short)0, a0, false, false);
    a1 = __builtin_amdgcn_wmma_f32_16x16x4_f32(false, a, false, b1, (short)0, a1, false, false);
    a2 = __builtin_amdgcn_wmma_f32_16x16x4_f32(false, a, false, b2, (short)0, a2, false, false);
    a3 = __builtin_amdgcn_wmma_f32_16x16x4_f32(false, a, false, b3, (short)0, a3, false, false);
  }

  const int rbase = blockIdx.x * 16 + ((lane >> 4) << 3);
#pragma unroll
  for (int t = 0; t < 4; ++t) {
    const int col = cbase + t * 16 + l15;
    if (col >= C) continue;
    const v8f& acc = (t == 0) ? a0 : (t == 1) ? a1 : (t == 2) ? a2 : a3;
    const float bv = bias[col];
#pragma unroll
    for (int r = 0; r < 8; ++r) {
      float v = acc[r] + bv;
      if (ACT == 1)      v = fmaxf(v, 0.0f);
      else if (ACT == 2) v = (v > 0.0f) ? v : 0.2f * v;
      const int orow = rbase + r;
      const long prow = obs ? ((long)(orow / obr) * obs + (orow % obr)) : orow;
      out[prow * (long)C + col] = v;
    }
  }
}

// ---------------------------------------------------------------------------
// Farthest point sampling: one block per batch; (value, min-index) argmax
// reduction reproduces jnp.argmax tie-breaking.
// ---------------------------------------------------------------------------
__global__ void fps_kernel(const float* __restrict__ xyz, int* __restrict__ fidx,
                           int N, int npoint) {
  __shared__ float mind[2048];
  __shared__ float rv[256];
  __shared__ int   ri[256];
  __shared__ int   s_last;
  const int b = blockIdx.x, tid = threadIdx.x;
  const float* X = xyz + (long)b * N * 3;
  for (int i = tid; i < N; i += 256) mind[i] = 1.0e10f;
  if (tid == 0) s_last = 0;
  __syncthreads();
  for (int it = 0; it < npoint; ++it) {
    const int last = s_last;
    if (tid == 0) fidx[b * npoint + it] = last;
    const float lx = X[last * 3 + 0], ly = X[last * 3 + 1], lz = X[last * 3 + 2];
    float bestv = -1.0f; int besti = 0;
    for (int i = tid; i < N; i += 256) {
      const float dx = X[i * 3 + 0] - lx;
      const float dy = X[i * 3 + 1] - ly;
      const float dz = X[i * 3 + 2] - lz;
      const float m = fminf(mind[i], dx * dx + dy * dy + dz * dz);
      mind[i] = m;
      if (m > bestv) { bestv = m; besti = i; }  // strict >: lowest index on tie
    }
    rv[tid] = bestv; ri[tid] = besti;
    __syncthreads();
    for (int s = 128; s > 0; s >>= 1) {
      if (tid < s) {
        const float v2 = rv[tid + s]; const int i2 = ri[tid + s];
        if (v2 > rv[tid] || (v2 == rv[tid] && i2 < ri[tid])) { rv[tid] = v2; ri[tid] = i2; }
      }
      __syncthreads();
    }
    if (tid == 0) s_last = ri[0];
    __syncthreads();
  }
}

__global__ void gather3_kernel(const float* __restrict__ xyz,
                               const int* __restrict__ fidx,
                               float* __restrict__ out, int N, int M, int total) {
  const int t = blockIdx.x * blockDim.x + threadIdx.x;
  if (t >= total) return;
  const int b = t / M;
  const int p = fidx[t];
  out[t * 3 + 0] = xyz[((long)b * N + p) * 3 + 0];
  out[t * 3 + 1] = xyz[((long)b * N + p) * 3 + 1];
  out[t * 3 + 2] = xyz[((long)b * N + p) * 3 + 2];
}

// Ball query with reference stable-argsort-of-mask semantics.
__global__ void ballq_kernel(const float* __restrict__ xyz,
                             const float* __restrict__ newxyz,
                             int* __restrict__ idx, int N, int M, float r2,
                             int total) {
  const int t = blockIdx.x * blockDim.x + threadIdx.x;
  if (t >= total) return;
  const int b = t / M;
  const float* X = xyz + (long)b * N * 3;
  const float nx = newxyz[t * 3 + 0], ny = newxyz[t * 3 + 1], nz = newxyz[t * 3 + 2];
  int* out = idx + (long)t * 32;
  int cnt = 0;
  for (int n = 0; n < N && cnt < 32; ++n) {
    const float dx = X[n * 3 + 0] - nx;
    const float dy = X[n * 3 + 1] - ny;
    const float dz = X[n * 3 + 2] - nz;
    if (dx * dx + dy * dy + dz * dz <= r2) out[cnt++] = n;
  }
  const int pad = (cnt > 0) ? out[0] : 0;
  for (int s = cnt; s < 32; ++s) out[s] = pad;
}

// 3-NN + inverse-distance weights (lowest index wins ties, like top_k).
__global__ void knn3_kernel(const float* __restrict__ xyz1,
                            const float* __restrict__ xyz2,
                            int* __restrict__ nidx, float* __restrict__ nw,
                            int N, int M2, int total) {
  const int t = blockIdx.x * blockDim.x + threadIdx.x;
  if (t >= total) return;
  const int b = t / N;
  const float px = xyz1[t * 3 + 0], py = xyz1[t * 3 + 1], pz = xyz1[t * 3 + 2];
  const float* X2 = xyz2 + (long)b * M2 * 3;
  float d0 = 1.0e30f, d1 = 1.0e30f, d2 = 1.0e30f;
  int i0 = 0, i1 = 0, i2 = 0;
  for (int m = 0; m < M2; ++m) {
    const float dx = X2[m * 3 + 0] - px;
    const float dy = X2[m * 3 + 1] - py;
    const float dz = X2[m * 3 + 2] - pz;
    const float d = dx * dx + dy * dy + dz * dz;
    if (d < d0)      { d2 = d1; i2 = i1; d1 = d0; i1 = i0; d0 = d; i0 = m; }
    else if (d < d1) { d2 = d1; i2 = i1; d1 = d;  i1 = m; }
    else if (d < d2) { d2 = d;  i2 = m; }
  }
  d0 = fmaxf(d0, 1e-10f); d1 = fmaxf(d1, 1e-10f); d2 = fmaxf(d2, 1e-10f);
  float w0 = 1.0f / d0, w1 = 1.0f / d1, w2 = 1.0f / d2;
  const float s = w0 + w1 + w2;
  nidx[t * 3 + 0] = i0; nidx[t * 3 + 1] = i1; nidx[t * 3 + 2] = i2;
  nw[t * 3 + 0] = w0 / s; nw[t * 3 + 1] = w1 / s; nw[t * 3 + 2] = w2 / s;
}

__global__ void interp_kernel(const float* __restrict__ pts2,
                              const int* __restrict__ nidx,
                              const float* __restrict__ nw,
                              float* __restrict__ out,
                              int C, int M2, int totalRows, int N) {
  const long t = (long)blockIdx.x * blockDim.x + threadIdx.x;
  if (t >= (long)totalRows * C) return;
  const int row = (int)(t / C), c = (int)(t % C);
  const int b = row / N;
  const int* id = nidx + (long)row * 3;
  const float* w = nw + (long)row * 3;
  const float* P = pts2 + (long)b * M2 * C;
  out[t] = w[0] * P[(long)id[0] * C + c] +
           w[1] * P[(long)id[1] * C + c] +
           w[2] * P[(long)id[2] * C + c];
}

// concat [up4(64), up3(64), up2(64), l1p(64), l0_xyz(3)] into stride-260 rows
// (column 259 zeroed so downstream GEMM K-loops are guard-free).
__global__ void concat260_kernel(const float* __restrict__ u4,
                                 const float* __restrict__ u3,
                                 const float* __restrict__ u2,
                                 const float* __restrict__ l1p,
                                 const float* __restrict__ l0,
                                 float* __restrict__ out, int totalRows) {
  const long t = (long)blockIdx.x * blockDim.x + threadIdx.x;
  if (t >= (long)totalRows * 260) return;
  const int row = (int)(t / 260), c = (int)(t % 260);
  float v = 0.0f;
  if (c < 64)       v = u4[(long)row * 64 + c];
  else if (c < 128) v = u3[(long)row * 64 + (c - 64)];
  else if (c < 192) v = u2[(long)row * 64 + (c - 128)];
  else if (c < 256) v = l1p[(long)row * 64 + (c - 192)];
  else if (c < 259) v = l0[(long)row * 3 + (c - 256)];
  out[t] = v;
}

// ---------------------------------------------------------------------------

extern "C" void kernel_launch(void* const* d_in, const int* in_sizes, int n_in,
                              void* d_out, int out_size, void* d_ws, size_t ws_size,
                              hipStream_t stream) {
  (void)in_sizes; (void)n_in; (void)out_size; (void)ws_size;
  const int B = 8, N = 2048;
  const float* inputs = (const float*)d_in[0];
  auto P = [&](int i) { return (const float*)d_in[i]; };

  char* ws = (char*)d_ws;
  size_t off = 0;
  auto alloc = [&](size_t bytes) -> void* {
    void* p = ws + off;
    off += (bytes + 255) & ~(size_t)255;
    return p;
  };

  int*   fidx1  = (int*)alloc((size_t)B * 2048 * 4);
  float* l1xyz  = (float*)alloc((size_t)B * 2048 * 3 * 4);
  int*   gidx1  = (int*)alloc((size_t)B * 2048 * 32 * 4);
  float* l1p    = (float*)alloc((size_t)B * 2048 * 64 * 4);
  int*   fidx2  = (int*)alloc((size_t)B * 1024 * 4);
  float* l2xyz  = (float*)alloc((size_t)B * 1024 * 3 * 4);
  int*   gidx2  = (int*)alloc((size_t)B * 1024 * 32 * 4);
  float* l2p    = (float*)alloc((size_t)B * 1024 * 128 * 4);
  int*   fidx3  = (int*)alloc((size_t)B * 512 * 4);
  float* l3xyz  = (float*)alloc((size_t)B * 512 * 3 * 4);
  int*   gidx3  = (int*)alloc((size_t)B * 512 * 32 * 4);
  float* l3p    = (float*)alloc((size_t)B * 512 * 256 * 4);
  int*   fidx4  = (int*)alloc((size_t)B * 256 * 4);
  float* l4xyz  = (float*)alloc((size_t)B * 256 * 3 * 4);
  int*   gidx4  = (int*)alloc((size_t)B * 256 * 32 * 4);
  float* l4p    = (float*)alloc((size_t)B * 256 * 512 * 4);
  int*   nnidx  = (int*)alloc((size_t)B * 2048 * 3 * 4);
  float* nnw    = (float*)alloc((size_t)B * 2048 * 3 * 4);
  float* interpB= (float*)alloc((size_t)B * 2048 * 512 * 4);
  float* up4    = (float*)alloc((size_t)B * 2048 * 64 * 4);
  float* up3    = (float*)alloc((size_t)B * 2048 * 64 * 4);
  float* up2    = (float*)alloc((size_t)B * 2048 * 64 * 4);
  float* ccat   = (float*)alloc((size_t)B * 2048 * 260 * 4);
  float* brmid  = (float*)alloc((size_t)B * 2048 * 256 * 4);
  float* net    = (float*)alloc((size_t)B * 8192 * 128 * 4);
  float* c1buf  = (float*)alloc((size_t)B * 8192 * 64 * 4);

  // --- pack all weights into zero-padded (K/4, C, 4) layout ---------------
  // weight input index -> (K, C); biases are at idx+1.
  struct WInfo { int idx, K, C; };
  const WInfo winfo[] = {
    {1, 3, 32},   {3, 32, 32},   {5, 32, 64},      // sa1
    {7, 67, 64},  {9, 64, 64},   {11, 64, 128},    // sa2
    {13, 131, 128},{15, 128, 128},{17, 128, 256},  // sa3
    {19, 259, 256},{21, 256, 256},{23, 256, 512},  // sa4
    {25, 512, 64}, {27, 256, 64}, {29, 128, 64},   // fp1..3
    {31, 259, 256},{33, 256, 128},                 // up br0
    {35, 259, 256},{37, 256, 128},                 // up br1
    {39, 259, 256},{41, 256, 128},                 // up br2
    {43, 259, 256},{45, 256, 128},                 // up br3
    {47, 128, 64}, {49, 64, 3},                    // final1, final2
  };
  float* pw[51] = {};
  for (const WInfo& w : winfo) {
    const int KP4 = (w.K + 3) & ~3;
    const int tot = KP4 * w.C;
    float* dst = (float*)alloc((size_t)tot * 4);
    pw[w.idx] = dst;
    pack_w_kernel<<<(tot + 255) / 256, 256, 0, stream>>>(P(w.idx), dst, w.K, w.C, tot);
  }

  // --- SA stages ----------------------------------------------------------
  const float R2[4] = {0.05f * 0.05f, 0.1f * 0.1f, 0.2f * 0.2f, 0.3f * 0.3f};
  auto smem_sa = [](int cin, int c1, int c2, int c3) -> size_t {
    const int kp1 = (cin + 3) & ~3;
    const int sa = (((kp1 > c2) ? kp1 : c2) + 2) & ~1;
    const int sb = (((c1 > c3) ? c1 : c3) + 2) & ~1;
    return (size_t)32 * (sa + sb) * 4;
  };

  struct Lv { int Nl, M; float r2; const float* xyzp; int* fidx; float* nxyz; int* gidx; };
  const Lv lv[4] = {
    {2048, 2048, R2[0], inputs, fidx1, l1xyz, gidx1},
    {2048, 1024, R2[1], l1xyz,  fidx2, l2xyz, gidx2},
    {1024,  512, R2[2], l2xyz,  fidx3, l3xyz, gidx3},
    { 512,  256, R2[3], l3xyz,  fidx4, l4xyz, gidx4},
  };
  for (int s = 0; s < 4; ++s) {
    const Lv& c = lv[s];
    const int total = B * c.M;
    fps_kernel<<<B, 256, 0, stream>>>(c.xyzp, c.fidx, c.Nl, c.M);
    gather3_kernel<<<(total + 255) / 256, 256, 0, stream>>>(c.xyzp, c.fidx, c.nxyz, c.Nl, c.M, total);
    ballq_kernel<<<(total + 255) / 256, 256, 0, stream>>>(c.xyzp, c.nxyz, c.gidx, c.Nl, c.M, c.r2, total);
  }
  sa_fused_kernel<3, 32, 32, 64, 0><<<B * 2048, 64, smem_sa(3, 32, 32, 64), stream>>>(
      inputs, nullptr, l1xyz, gidx1, pw[1], P(2), pw[3], P(4), pw[5], P(6), l1p, 2048, 2048);
  sa_fused_kernel<67, 64, 64, 128, 64><<<B * 1024, 64, smem_sa(67, 64, 64, 128), stream>>>(
      l1xyz, l1p, l2xyz, gidx2, pw[7], P(8), pw[9], P(10), pw[11], P(12), l2p, 2048, 1024);
  sa_fused_kernel<131, 128, 128, 256, 128><<<B * 512, 64, smem_sa(131, 128, 128, 256), stream>>>(
      l2xyz, l2p, l3xyz, gidx3, pw[13], P(14), pw[15], P(16), pw[17], P(18), l3p, 1024, 512);
  sa_fused_kernel<259, 256, 256, 512, 256><<<B * 256, 64, smem_sa(259, 256, 256, 512), stream>>>(
      l3xyz, l3p, l4xyz, gidx4, pw[19], P(20), pw[21], P(22), pw[23], P(24), l4p, 512, 256);

  // --- feature propagation ------------------------------------------------
  const int rowsN = B * N;  // 16384
  struct FpCfg { const float* x2; const float* p2; int C2, M2, wi; float* out; };
  const FpCfg fp[3] = {
    {l4xyz, l4p, 512, 256, 25, up4},
    {l3xyz, l3p, 256, 512, 27, up3},
    {l2xyz, l2p, 128, 1024, 29, up2},
  };
  for (int s = 0; s < 3; ++s) {
    const FpCfg& c = fp[s];
    knn3_kernel<<<(rowsN + 255) / 256, 256, 0, stream>>>(inputs, c.x2, nnidx, nnw, N, c.M2, rowsN);
    const long tot = (long)rowsN * c.C2;
    interp_kernel<<<(int)((tot + 255) / 256), 256, 0, stream>>>(c.p2, nnidx, nnw, interpB, c.C2, c.M2, rowsN, N);
    dim3 g(rowsN / 16, 1);  // C=64 -> one 64-col block
    gemm_wmma_kernel<1><<<g, 32, 0, stream>>>(interpB, pw[c.wi], P(c.wi + 1), c.out,
                                              rowsN, c.C2, c.C2, 64, 1, 0);
  }

  // --- concat + 4 up-branches --------------------------------------------
  {
    const long tot = (long)rowsN * 260;
    concat260_kernel<<<(int)((tot + 255) / 256), 256, 0, stream>>>(up4, up3, up2, l1p, inputs, ccat, rowsN);
  }
  for (int br = 0; br < 4; ++br) {
    const int wi = 31 + br * 4;
    dim3 g1(rowsN / 16, 4);  // C=256
    gemm_wmma_kernel<1><<<g1, 32, 0, stream>>>(ccat, pw[wi], P(wi + 1), brmid,
                                               rowsN, 260, 260, 256, 1, 0);
    dim3 g2(rowsN / 16, 2);  // C=128, write into axis-1 concat layout
    gemm_wmma_kernel<1><<<g2, 32, 0, stream>>>(brmid, pw[wi + 2], P(wi + 3),
                                               net + (long)br * 2048 * 128,
                                               rowsN, 256, 256, 128, 2048, 8192);
  }

  // --- final head ---------------------------------------------------------
  const int rowsF = B * 8192;  // 65536
  {
    dim3 g(rowsF / 16, 1);  // C=64
    gemm_wmma_kernel<2><<<g, 32, 0, stream>>>(net, pw[47], P(48), c1buf,
                                              rowsF, 128, 128, 64, 1, 0);
  }
  {
    dim3 g(rowsF / 16, 1);  // C=3
    gemm_wmma_kernel<0><<<g, 32, 0, stream>>>(c1buf, pw[49], P(50), (float*)d_out,
                                              rowsF, 64, 64, 3, 1, 0);
  }
}